// SAGE_30056181137772
// MI455X (gfx1250) — compile-verified
//
#include <hip/hip_runtime.h>
#include <hip/hip_bf16.h>

typedef __attribute__((ext_vector_type(2))) float v2f;
typedef __attribute__((ext_vector_type(4))) float v4f;
typedef __attribute__((ext_vector_type(8))) float v8f;

// ---------------------------------------------------------------------------
// C[n x fo] = H[n x fi] @ W[fi x fo] (+ bias), via V_WMMA_F32_16X16X4_F32.
// One 16-row M tile per block; 8 waves per block each own N tiles (stride 8).
// A layout (32-bit 16x4): lanes 0-15 -> K=k+0,k+1 ; lanes 16-31 -> K=k+2,k+3.
// C/D layout: VGPR r -> row m0 + r + 8*(lane>=16), col n0 + (lane&15).
// Requires n % 16 == 0 and fi % 4 == 0 (true for all three layers).
// ---------------------------------------------------------------------------
__global__ __launch_bounds__(256) void sage_gemm_wmma(
    const float* __restrict__ H, const float* __restrict__ W,
    const float* __restrict__ bias, float* __restrict__ C,
    int n, int fi, int fo)
{
    const int lane = threadIdx.x & 31;
    const int wave = threadIdx.x >> 5;
    const int half = lane >> 4;      // 0: lanes 0-15, 1: lanes 16-31
    const int l16  = lane & 15;
    const int m0   = blockIdx.x << 4;
    const int tiles_n = (fo + 15) >> 4;

    const float* __restrict__ Hrow = H + (size_t)(m0 + l16) * fi;

    for (int nt = wave; nt < tiles_n; nt += 8) {
        const int n0   = nt << 4;
        const int bcol = n0 + l16;
        const bool colok = (bcol < fo);
        const int  bc    = colok ? bcol : 0;   // clamp OOB lanes (fo=47 tail)

        v8f acc = {};
        for (int k = 0; k < fi; k += 4) {
            v2f a;
            const float* __restrict__ ap = Hrow + k + 2 * half;
            a.x = ap[0];
            a.y = ap[1];
            v2f b;
            const float* __restrict__ bp = W + (size_t)(k + 2 * half) * fo + bc;
            b.x = bp[0];
            b.y = bp[fo];
            // D = A(16x4) * B(4x16) + C, fp32 in / fp32 accumulate
            acc = __builtin_amdgcn_wmma_f32_16x16x4_f32(
                false, a, false, b, (short)0, acc, false, false);
        }

        if (colok) {
            const float bb = bias ? bias[bcol] : 0.0f;
            const int rbase = m0 + 8 * half;
#pragma unroll
            for (int r = 0; r < 8; ++r) {
                C[(size_t)(rbase + r) * fo + bcol] = acc[r] + bb;
            }
        }
    }
}

// ---------------------------------------------------------------------------
// Graph preprocessing: in-degree count -> inv_deg, rowptr (scan), CSR fill,
// per-row sort (restores determinism after atomic bucket fill).
// ---------------------------------------------------------------------------
__global__ void zero_i32(int* __restrict__ p, int n) {
    int i = blockIdx.x * blockDim.x + threadIdx.x;
    if (i < n) p[i] = 0;
}

__global__ void count_deg_i32(const int* __restrict__ dst, int* __restrict__ deg,
                              int n_edges) {
    int i = blockIdx.x * blockDim.x + threadIdx.x;
    if (i < n_edges) atomicAdd(&deg[dst[i]], 1);
}

__global__ void invdeg_from_i32(const int* __restrict__ deg,
                                float* __restrict__ inv_deg, int n) {
    int i = blockIdx.x * blockDim.x + threadIdx.x;
    if (i < n) inv_deg[i] = 1.0f / fmaxf((float)deg[i], 1.0f);
}

// Single-block exclusive scan over n degrees -> rowptr[0..n] (rowptr[n] = E).
__global__ __launch_bounds__(1024) void scan_rowptr(
    const int* __restrict__ deg, int* __restrict__ rowptr, int n)
{
    __shared__ int smem[1024];
    __shared__ int carry;
    if (threadIdx.x == 0) carry = 0;
    __syncthreads();
    for (int base = 0; base < n; base += 1024) {
        const int i = base + (int)threadIdx.x;
        const int v = (i < n) ? deg[i] : 0;
        smem[threadIdx.x] = v;
        __syncthreads();
        for (int off = 1; off < 1024; off <<= 1) {
            int t = 0;
            if ((int)threadIdx.x >= off) t = smem[threadIdx.x - off];
            __syncthreads();
            smem[threadIdx.x] += t;
            __syncthreads();
        }
        if (i < n) rowptr[i] = carry + smem[threadIdx.x] - v;
        __syncthreads();                 // everyone read carry before update
        if (threadIdx.x == 0) carry += smem[1023];
        __syncthreads();
    }
    if (threadIdx.x == 0) rowptr[n] = carry;
}

__global__ void fill_csr(const int* __restrict__ src, const int* __restrict__ dst,
                         const int* __restrict__ rowptr, int* __restrict__ cursor,
                         int* __restrict__ csr_src, int n_edges) {
    int i = blockIdx.x * blockDim.x + threadIdx.x;
    if (i < n_edges) {
        const int d = dst[i];
        const int pos = atomicAdd(&cursor[d], 1);
        csr_src[rowptr[d] + pos] = src[i];
    }
}

// Insertion-sort each CSR row by src id -> deterministic summation order.
__global__ void sort_rows(const int* __restrict__ rowptr,
                          int* __restrict__ csr_src, int n) {
    int i = blockIdx.x * blockDim.x + threadIdx.x;
    if (i >= n) return;
    const int b = rowptr[i], e = rowptr[i + 1];
    for (int p = b + 1; p < e; ++p) {
        const int v = csr_src[p];
        int q = p - 1;
        while (q >= b && csr_src[q] > v) { csr_src[q + 1] = csr_src[q]; --q; }
        csr_src[q + 1] = v;
    }
}

// ---------------------------------------------------------------------------
// Gather-aggregate: out[d] += (sum_{s in N(d)} msg[s]) * inv_deg[d]; opt ReLU.
// One wave per node; lanes cover features; atomic-free, fully coalesced.
// ---------------------------------------------------------------------------
__global__ __launch_bounds__(256) void sage_gather(
    const float* __restrict__ msg, const int* __restrict__ rowptr,
    const int* __restrict__ csr_src, const float* __restrict__ inv_deg,
    float* __restrict__ out, int n, int fo, int do_relu)
{
    const int lane = threadIdx.x & 31;
    const int node = blockIdx.x * 8 + (threadIdx.x >> 5);
    if (node >= n) return;
    const int beg = rowptr[node], end = rowptr[node + 1];
    const float w = inv_deg[node];

    if (fo == 256) {                      // vectorized path: 2 x b128 per lane
        v4f acc0 = {}, acc1 = {};
        for (int e = beg; e < end; ++e) {
            const v4f* __restrict__ m4 =
                (const v4f*)(msg + (size_t)csr_src[e] * 256);
            acc0 += m4[lane];
            acc1 += m4[lane + 32];
        }
        v4f* __restrict__ o4 = (v4f*)(out + (size_t)node * 256);
        v4f s0 = o4[lane]      + acc0 * w;
        v4f s1 = o4[lane + 32] + acc1 * w;
        if (do_relu) {
#pragma unroll
            for (int c = 0; c < 4; ++c) {
                s0[c] = fmaxf(s0[c], 0.0f);
                s1[c] = fmaxf(s1[c], 0.0f);
            }
        }
        o4[lane]      = s0;
        o4[lane + 32] = s1;
    } else {                              // generic path (fo = 47)
        float acc[8];
        const int nf = (fo + 31) >> 5;
#pragma unroll
        for (int j = 0; j < 8; ++j) acc[j] = 0.0f;
        for (int e = beg; e < end; ++e) {
            const float* __restrict__ mrow = msg + (size_t)csr_src[e] * fo;
            for (int j = 0; j < nf; ++j) {
                const int f = lane + (j << 5);
                if (f < fo) acc[j] += mrow[f];
            }
        }
        float* __restrict__ orow = out + (size_t)node * fo;
        for (int j = 0; j < nf; ++j) {
            const int f = lane + (j << 5);
            if (f < fo) {
                float v = orow[f] + acc[j] * w;
                orow[f] = do_relu ? fmaxf(v, 0.0f) : v;
            }
        }
    }
}

// ---------------------------------------------------------------------------
extern "C" void kernel_launch(void* const* d_in, const int* in_sizes, int n_in,
                              void* d_out, int out_size, void* d_ws, size_t ws_size,
                              hipStream_t stream) {
    const float* x   = (const float*)d_in[0];
    const int*   src = (const int*)  d_in[1];
    const int*   dst = (const int*)  d_in[2];
    const float* Ws0 = (const float*)d_in[3];
    const float* Wn0 = (const float*)d_in[4];
    const float* b0  = (const float*)d_in[5];
    const float* Ws1 = (const float*)d_in[6];
    const float* Wn1 = (const float*)d_in[7];
    const float* b1  = (const float*)d_in[8];
    const float* Ws2 = (const float*)d_in[9];
    const float* Wn2 = (const float*)d_in[10];
    const float* b2  = (const float*)d_in[11];

    const int N  = 50000;
    const int E  = in_sizes[1];     // 400000
    const int FI = in_sizes[0] / N; // 988
    const int HD = 256;
    const int NC = 47;

    // Workspace layout (16B-aligned buffers first for v4f loads).
    char*  wsb     = (char*)d_ws;
    float* hA      = (float*)wsb;                       // N x 256
    float* tmsg    = hA   + (size_t)N * HD;             // N x 256
    float* hB      = tmsg + (size_t)N * HD;             // N x 256
    float* inv_deg = hB   + (size_t)N * HD;             // N
    int*   deg_i   = (int*)(inv_deg + N);               // N
    int*   cursor  = deg_i + N;                         // N
    int*   csr     = cursor + N;                        // E
    int*   rowptr  = csr + E;                           // N+1
    float* out     = (float*)d_out;                     // N x 47

    // --- graph preprocessing (once; reused by all 3 layers) ---
    zero_i32       <<<(N + 255) / 256, 256, 0, stream>>>(deg_i, N);
    count_deg_i32  <<<(E + 255) / 256, 256, 0, stream>>>(dst, deg_i, E);
    invdeg_from_i32<<<(N + 255) / 256, 256, 0, stream>>>(deg_i, inv_deg, N);
    scan_rowptr    <<<1, 1024, 0, stream>>>(deg_i, rowptr, N);
    zero_i32       <<<(N + 255) / 256, 256, 0, stream>>>(cursor, N);
    fill_csr       <<<(E + 255) / 256, 256, 0, stream>>>(src, dst, rowptr, cursor, csr, E);
    sort_rows      <<<(N + 255) / 256, 256, 0, stream>>>(rowptr, csr, N);

    const dim3 ggrid(N / 16);    // 3125 M-tiles, 8 waves/block over N tiles
    const dim3 agrid(N / 8);     // 6250 blocks, 1 wave per node

    // --- layer 0: 988 -> 256, ReLU ---
    sage_gemm_wmma<<<ggrid, 256, 0, stream>>>(x, Ws0, b0,      hA,   N, FI, HD);
    sage_gemm_wmma<<<ggrid, 256, 0, stream>>>(x, Wn0, nullptr, tmsg, N, FI, HD);
    sage_gather   <<<agrid, 256, 0, stream>>>(tmsg, rowptr, csr, inv_deg, hA, N, HD, 1);

    // --- layer 1: 256 -> 256, ReLU ---
    sage_gemm_wmma<<<ggrid, 256, 0, stream>>>(hA, Ws1, b1,      hB,   N, HD, HD);
    sage_gemm_wmma<<<ggrid, 256, 0, stream>>>(hA, Wn1, nullptr, tmsg, N, HD, HD);
    sage_gather   <<<agrid, 256, 0, stream>>>(tmsg, rowptr, csr, inv_deg, hB, N, HD, 1);

    // --- layer 2: 256 -> 47, no ReLU, straight into d_out ---
    sage_gemm_wmma<<<ggrid, 256, 0, stream>>>(hB, Ws2, b2,      out,  N, HD, NC);
    sage_gemm_wmma<<<ggrid, 256, 0, stream>>>(hB, Wn2, nullptr, tmsg, N, HD, NC);
    sage_gather   <<<agrid, 256, 0, stream>>>(tmsg, rowptr, csr, inv_deg, out, N, NC, 0);
}